// EquilibriumClass_32444182954344
// MI455X (gfx1250) — compile-verified
//
#include <hip/hip_runtime.h>

// Chain relaxation: B=256 chains, N=1026 nodes (2D), 64 damped-Euler steps.
// One workgroup (1024 threads = 32 wave32s) per chain; all state lives in LDS
// for all 64 steps. gfx1250 paths: async global->LDS load of the initial
// state, async LDS->global store of the result, split workgroup barriers.

#define NN      1026      // nodes per chain
#define HH      1024      // angles per chain
#define NEDGE   1025      // edges per chain
#define NCOORD  2052      // 2*NN
#define NCHUNK  513       // 16-byte chunks per chain row (2052 floats)
#define BLK     1024      // 32 wave32s: phase2 is a single pass
#define LREST   1.0f
#define DTS     0.001f
#define GAM     2.0f
#define NSTEPS  64

__global__ __launch_bounds__(BLK) void chain_sim_kernel(
    const float* __restrict__ pos0,     // (B, N, 2)
    const float* __restrict__ tip,      // (B, 2)
    const float* __restrict__ thetas,   // (H,)
    const int*   __restrict__ buckle,   // (H, 4)
    const float* __restrict__ p_kstiff,
    const float* __restrict__ p_ksoft,
    const float* __restrict__ p_kstr,
    float*       __restrict__ out)      // (B, NC)
{
    __shared__ __align__(16) float2 s_pos[NN];
    __shared__ float2 s_vel[NN];
    __shared__ float2 s_Fe[NEDGE];              // stretch force along each edge
    __shared__ __align__(16) float4 s_AB[HH];   // (a.x,a.y,b.x,b.y) per angle
    __shared__ float  s_TH[HH];                 // theta_ss
    __shared__ float  s_NP[HH];                 // count of buckle==1 over S=4

    const int tid = threadIdx.x;
    const int b   = blockIdx.x;

    const float k_stiff = p_kstiff[0];
    const float k_soft  = p_ksoft[0];
    const float k_str   = p_kstr[0];

    // ---- async DMA: pos0 row (8208 B) -> LDS, 16 B per lane per issue ----
    {
        const unsigned lds_base = (unsigned)(unsigned long long)(&s_pos[0]);
        const unsigned long long gsrc =
            (unsigned long long)(pos0) + (unsigned long long)b * (NCOORD * 4ull);
        for (int c = tid; c < NCHUNK; c += BLK) {
            unsigned laddr = lds_base + (unsigned)c * 16u;
            unsigned long long gaddr = gsrc + (unsigned long long)c * 16ull;
            asm volatile("global_load_async_to_lds_b128 %0, %1, off"
                         :: "v"(laddr), "v"(gaddr) : "memory");
        }
        asm volatile("s_wait_asynccnt 0" ::: "memory");
    }

    // ---- per-angle constants + zero velocities ----
    for (int i = tid; i < NN; i += BLK) s_vel[i] = make_float2(0.0f, 0.0f);
    for (int k = tid; k < HH; k += BLK) {
        s_TH[k] = thetas[k];
        const int* bp = buckle + k * 4;
        s_NP[k] = (float)(bp[0] + bp[1] + bp[2] + bp[3]);
    }
    __syncthreads();
    if (tid == 0) {
        s_pos[NN - 1] = make_float2(tip[2 * b], tip[2 * b + 1]);  // pin tip
    }
    __syncthreads();

    // ---- 64 sequential dynamics steps, entirely in LDS ----
    for (int step = 0; step < NSTEPS; ++step) {
        // Phase 1: per-edge stretch force + per-angle bending terms (read pos)
        for (int i = tid; i < NEDGE; i += BLK) {
            float2 p0 = s_pos[i];
            float2 p1 = s_pos[i + 1];
            float dx = p1.x - p0.x, dy = p1.y - p0.y;
            float d2  = dx * dx + dy * dy;
            float len = sqrtf(d2);
            float coef = k_str * (len - LREST) / len;
            s_Fe[i] = make_float2(coef * dx, coef * dy);

            if (i < HH) {  // angle k=i at vertex i+1: v1 = edge i, v2 = edge i+1
                float2 p2 = s_pos[i + 2];
                float ex = p2.x - p1.x, ey = p2.y - p1.y;
                float crs = dx * ey - dy * ex;
                float dot = dx * ex + dy * ey;
                float th  = atan2f(crs, dot);
                float tss = s_TH[i];
                float np  = s_NP[i];
                // pm=+1 entries stiff iff th<tss; pm=-1 entries stiff iff th>-tss
                float kp = (th <  tss) ? k_stiff : k_soft;
                float km = (th > -tss) ? k_stiff : k_soft;
                float K  = np * kp + (4.0f - np) * km;
                float M  = K * (th - tss);
                float inv1 = 1.0f / d2;
                float inv2 = 1.0f / (ex * ex + ey * ey);
                s_AB[i] = make_float4(-M * dy * inv1, M * dx * inv1,
                                      -M * ey * inv2, M * ex * inv2);
            }
        }
        __syncthreads();

        // Phase 2: gather forces, update free nodes 2..1024 (single pass)
        for (int i = tid + 2; i <= NN - 2; i += BLK) {
            float4 ab1 = s_AB[i - 1];          // a_{i-1}, b_{i-1}  (one b128)
            float bx2 = s_AB[i - 2].z;         // b_{i-2}           (one b64)
            float by2 = s_AB[i - 2].w;
            float fx = s_Fe[i].x - s_Fe[i - 1].x + ab1.x + ab1.z - bx2;
            float fy = s_Fe[i].y - s_Fe[i - 1].y + ab1.y + ab1.w - by2;
            if (i <= HH - 1) {                 // a_i               (one b64)
                fx -= s_AB[i].x;
                fy -= s_AB[i].y;
            }

            float2 v = s_vel[i];
            v.x += DTS * (fx - GAM * v.x);
            v.y += DTS * (fy - GAM * v.y);
            s_vel[i] = v;

            float2 p = s_pos[i];
            p.x += DTS * v.x;
            p.y += DTS * v.y;
            s_pos[i] = p;
        }
        __syncthreads();
    }

    // ---- async DMA: LDS -> out row ----
    {
        const unsigned lds_base = (unsigned)(unsigned long long)(&s_pos[0]);
        const unsigned long long gdst =
            (unsigned long long)(out) + (unsigned long long)b * (NCOORD * 4ull);
        for (int c = tid; c < NCHUNK; c += BLK) {
            unsigned laddr = lds_base + (unsigned)c * 16u;
            unsigned long long gaddr = gdst + (unsigned long long)c * 16ull;
            asm volatile("global_store_async_from_lds_b128 %0, %1, off"
                         :: "v"(gaddr), "v"(laddr) : "memory");
        }
        asm volatile("s_wait_asynccnt 0" ::: "memory");
    }
}

extern "C" void kernel_launch(void* const* d_in, const int* in_sizes, int n_in,
                              void* d_out, int out_size, void* d_ws, size_t ws_size,
                              hipStream_t stream) {
    const float* pos0    = (const float*)d_in[0];
    const float* tip     = (const float*)d_in[1];
    const float* thetas  = (const float*)d_in[2];
    const int*   buckle  = (const int*)d_in[3];
    const float* kstiff  = (const float*)d_in[4];
    const float* ksoft   = (const float*)d_in[5];
    const float* kstr    = (const float*)d_in[6];

    const int B = in_sizes[0] / NCOORD;  // 256
    chain_sim_kernel<<<B, BLK, 0, stream>>>(pos0, tip, thetas, buckle,
                                            kstiff, ksoft, kstr, (float*)d_out);
}